// VoxelHashTable_15891378995271
// MI455X (gfx1250) — compile-verified
//
#include <hip/hip_runtime.h>
#include <stdint.h>

// VoxelHashTable query for MI455X (gfx1250, wave32).
// wave-per-(point,level), lane = feature dim (D=32 == wave32).
// Gathers 8 corner feature rows via GLOBAL_LOAD_ASYNC_TO_LDS_B128 (ASYNCcnt),
// double-buffered in LDS, trilinear blend reads via ds_load, coalesced store.

#define WAVES_PER_BLOCK 8
#define BLOCK_THREADS   256

#if __has_builtin(__builtin_amdgcn_ballot_w32)
#define BALLOT32(p) __builtin_amdgcn_ballot_w32(p)
#else
#define BALLOT32(p) ((unsigned)__ballot(p))
#endif

typedef float SmemT[WAVES_PER_BLOCK][2][8][32];

struct ItemState {
  float fx, fy, fz;
  unsigned m0, m1;     // per-lane ballot of vid>=0 for corners z=0 / z=1
  unsigned outoff;     // p*64 + lv*32
};

// Issue the async gather for one (point, level) item into the LDS tile at
// lds_off (1024 bytes: 8 rows x 32 floats). Returns the blend state.
__device__ __forceinline__ ItemState issue_item(
    long long item, int lane,
    const float* __restrict__ pts,
    const int* __restrict__ h2v0, const int* __restrict__ h2v1,
    unsigned long long feat0, unsigned long long feat1,
    unsigned lds_off)
{
  ItemState st;
  const int lv = (int)(item & 1);
  const int p  = (int)(item >> 1);
  const float res = lv ? 0.12f : 0.24f;
  const int* __restrict__ h2v = lv ? h2v1 : h2v0;
  const unsigned long long fbase = lv ? feat1 : feat0;

  // Point coords (same address across lanes -> broadcast load).
  const float px = pts[3 * p + 0];
  const float py = pts[3 * p + 1];
  const float pz = pts[3 * p + 2];
  // Exact division to match reference voxel selection.
  const float qx = (px - (-2.6f)) / res;
  const float qy = (py - (-8.1f)) / res;
  const float qz = (pz -  0.0f ) / res;
  const float bxf = floorf(qx), byf = floorf(qy), bzf = floorf(qz);
  st.fx = qx - bxf; st.fy = qy - byf; st.fz = qz - bzf;
  const int bx = (int)bxf, by = (int)byf, bz = (int)bzf;

  // Lane l handles corner c0 = l>>3 (z=0) and c0+4 (z=1), 16B chunk l&7.
  const int c0 = lane >> 3;
  const int ox = c0 & 1, oy = (c0 >> 1) & 1;
  // uint32 wraparound hash == int64 hash mod 2^21 (2^21 divides 2^32).
  const unsigned s = (unsigned)(bx + ox) * 73856093u
                   + (unsigned)(by + oy) * 19349669u
                   + (unsigned)(bz)      * 83492791u;
  const unsigned hv0 = s & 0x1FFFFFu;
  const unsigned hv1 = (s + 83492791u) & 0x1FFFFFu;
  // Low dword of the int64 table entry: preserves -1 and small indices.
  const int vid0 = h2v[(size_t)hv0 * 2];
  const int vid1 = h2v[(size_t)hv1 * 2];
  st.m0 = BALLOT32(vid0 >= 0);
  st.m1 = BALLOT32(vid1 >= 0);

  const unsigned chunk = (unsigned)(lane & 7) * 16u;
  const unsigned voffA = (unsigned)(vid0 < 0 ? 0 : vid0) * 128u + chunk;
  const unsigned voffB = (unsigned)(vid1 < 0 ? 0 : vid1) * 128u + chunk;
  const unsigned ldsA  = lds_off + (unsigned)lane * 16u;   // rows 0..3
  const unsigned ldsB  = ldsA + 512u;                      // rows 4..7

  // WAR guard: prior blend's ds reads of this buffer must retire first.
  asm volatile("s_wait_dscnt 0x0" ::: "memory");
  // Per-lane gather: 32 lanes x 16B = 512B per instruction, ASYNCcnt-tracked.
  asm volatile("global_load_async_to_lds_b128 %0, %1, %2"
               :: "v"(ldsA), "v"(voffA), "s"(fbase) : "memory");
  asm volatile("global_load_async_to_lds_b128 %0, %1, %2"
               :: "v"(ldsB), "v"(voffB), "s"(fbase) : "memory");

  st.outoff = (unsigned)p * 64u + (unsigned)lv * 32u;
  return st;
}

// Blend from the LDS tile smem[wid][cur]; direct shared-array indexing keeps
// addrspace(3) so these are ds_load_b32 (bank-conflict-free, lane=dim).
__device__ __forceinline__ void process_item(
    const ItemState& st, int lane, int wid, int cur, SmemT& smem,
    float* __restrict__ out)
{
  const float fx = st.fx, fy = st.fy, fz = st.fz;
  const float gx = 1.0f - fx, gy = 1.0f - fy, gz = 1.0f - fz;
  float acc = 0.0f;
#pragma unroll
  for (int c = 0; c < 8; ++c) {
    float wx = (c & 1) ? fx : gx;
    float wy = (c & 2) ? fy : gy;
    float wz = (c & 4) ? fz : gz;
    float w = (wx * wy) * wz;
    const unsigned vbit = (c < 4) ? ((st.m0 >> (8 * c)) & 1u)
                                  : ((st.m1 >> (8 * (c - 4))) & 1u);
    w = vbit ? w : 0.0f;                          // invalid corner -> 0
    acc = fmaf(w, smem[wid][cur][c][lane], acc);  // ds_load_b32
  }
  out[st.outoff + (unsigned)lane] = acc;          // coalesced 128B store
}

__global__ void __launch_bounds__(BLOCK_THREADS)
voxel_hash_query_kernel(const float* __restrict__ pts,
                        const float* __restrict__ feat0,
                        const float* __restrict__ feat1,
                        const int* __restrict__ h2v0,
                        const int* __restrict__ h2v1,
                        float* __restrict__ out,
                        int npts)
{
  __shared__ SmemT smem;
  const int lane = (int)(threadIdx.x & 31u);
  const int wid  = (int)(threadIdx.x >> 5);
  const long long gw    = (long long)blockIdx.x * WAVES_PER_BLOCK + wid;
  const long long nw    = (long long)gridDim.x * WAVES_PER_BLOCK;
  const long long total = 2LL * (long long)npts;

  // Low 32 bits of the generic pointer == workgroup-relative LDS byte offset.
  // Used ONLY for the async-copy asm; blend reads index smem directly.
  const unsigned lds_base = (unsigned)(uintptr_t)&smem[wid][0][0][0];
  const unsigned long long f0 = (unsigned long long)(uintptr_t)feat0;
  const unsigned long long f1 = (unsigned long long)(uintptr_t)feat1;

  long long item = gw;
  if (item >= total) return;        // wave-uniform

  int cur = 0;
  ItemState s0 = issue_item(item, lane, pts, h2v0, h2v1, f0, f1, lds_base);
  long long nxt = item + nw;
  for (;;) {
    const bool have_next = (nxt < total);   // wave-uniform
    ItemState s1;
    if (have_next) {
      s1 = issue_item(nxt, lane, pts, h2v0, h2v1, f0, f1,
                      lds_base + (unsigned)(cur ^ 1) * 1024u);
      // 4 asyncs outstanding, in-order completion: <=2 left means current
      // buffer's two transfers have landed; next item's stay in flight.
      asm volatile("s_wait_asynccnt 0x2" ::: "memory");
    } else {
      asm volatile("s_wait_asynccnt 0x0" ::: "memory");
    }
    process_item(s0, lane, wid, cur, smem, out);
    if (!have_next) break;
    s0 = s1;
    cur ^= 1;
    nxt += nw;
  }
}

extern "C" void kernel_launch(void* const* d_in, const int* in_sizes, int n_in,
                              void* d_out, int out_size, void* d_ws, size_t ws_size,
                              hipStream_t stream) {
  const float* pts   = (const float*)d_in[0];
  const float* feat0 = (const float*)d_in[1];
  const float* feat1 = (const float*)d_in[2];
  const int*   h2v0  = (const int*)d_in[3];   // int64 table; low dwords read
  const int*   h2v1  = (const int*)d_in[4];
  float* out = (float*)d_out;
  const int npts = in_sizes[0] / 3;

  const long long total  = 2LL * (long long)npts;
  long long blocks = (total + WAVES_PER_BLOCK - 1) / WAVES_PER_BLOCK;
  if (blocks > 2048) blocks = 2048;
  if (blocks < 1) blocks = 1;

  hipLaunchKernelGGL(voxel_hash_query_kernel,
                     dim3((unsigned)blocks), dim3(BLOCK_THREADS), 0, stream,
                     pts, feat0, feat1, h2v0, h2v1, out, npts);
}